// GINConv_43980465111481
// MI455X (gfx1250) — compile-verified
//
#include <hip/hip_runtime.h>
#include <hip/hip_bf16.h>

typedef __attribute__((ext_vector_type(2))) float v2f;
typedef __attribute__((ext_vector_type(8))) float v8f;

#define NNODES 50000
#define NEDGES 1600000
#define DIM 128
#define BN_EPS 1e-5f

// ---------------------------------------------------------------------------
// Kernel 1: hbuf = (1 + eps) * x       (N*D floats, float4 per thread)
// ---------------------------------------------------------------------------
__global__ void k_init_h(const float* __restrict__ x, const float* __restrict__ eps,
                         float* __restrict__ hbuf) {
    size_t i = (size_t)blockIdx.x * blockDim.x + threadIdx.x;   // float4 index
    const float k = 1.0f + eps[0];
    float4 v = ((const float4*)x)[i];
    v.x *= k; v.y *= k; v.z *= k; v.w *= k;
    ((float4*)hbuf)[i] = v;
}

// ---------------------------------------------------------------------------
// Kernel 2: per-edge message + scatter-add.
// One wave (32 lanes) per edge; each lane owns 4 consecutive channels.
// hbuf[dst] += relu(x[src] + emb0 + emb1 + emb2)
// ---------------------------------------------------------------------------
__global__ void k_edge_scatter(const float* __restrict__ x,
                               const int* __restrict__ edge_attr,
                               const int* __restrict__ src,
                               const int* __restrict__ dst,
                               const float* __restrict__ bond_emb,
                               float* __restrict__ hbuf) {
    int e = blockIdx.x * 8 + (threadIdx.x >> 5);
    if (e >= NEDGES) return;
    int lane = threadIdx.x & 31;
    int col  = lane * 4;

    int s  = src[e];
    int d  = dst[e];
    int a0 = edge_attr[e * 3 + 0];
    int a1 = edge_attr[e * 3 + 1];
    int a2 = edge_attr[e * 3 + 2];

    float4 xv = *(const float4*)(x + (size_t)s * DIM + col);
    float4 e0 = *(const float4*)(bond_emb + (size_t)(0 * 5 + a0) * DIM + col);
    float4 e1 = *(const float4*)(bond_emb + (size_t)(1 * 5 + a1) * DIM + col);
    float4 e2 = *(const float4*)(bond_emb + (size_t)(2 * 5 + a2) * DIM + col);

    float m0 = fmaxf(xv.x + e0.x + e1.x + e2.x, 0.0f);
    float m1 = fmaxf(xv.y + e0.y + e1.y + e2.y, 0.0f);
    float m2 = fmaxf(xv.z + e0.z + e1.z + e2.z, 0.0f);
    float m3 = fmaxf(xv.w + e0.w + e1.w + e2.w, 0.0f);

    float* o = hbuf + (size_t)d * DIM + col;
    unsafeAtomicAdd(o + 0, m0);
    unsafeAtomicAdd(o + 1, m1);
    unsafeAtomicAdd(o + 2, m2);
    unsafeAtomicAdd(o + 3, m3);
}

// ---------------------------------------------------------------------------
// Kernel 3: zero the 256-float stats region (colsum[128] + colsumsq[128])
// ---------------------------------------------------------------------------
__global__ void k_zero_stats(float* __restrict__ stats) {
    stats[threadIdx.x] = 0.0f;
}

// ---------------------------------------------------------------------------
// Kernel 4: h1 = hbuf @ W1 + b1 via V_WMMA_F32_16X16X4_F32.
// One wave per 16x16 output tile. Fused column sum/sumsq for BatchNorm.
// A layout (f32 16x4): lanes 0-15 row M=l, K={k,k+1}; lanes 16-31 row M=l-16,
// K={k+2,k+3}. B layout mirrors on columns. C/D: lane holds col N=l%16,
// VGPR r -> row (l<16 ? r : r+8).
// ---------------------------------------------------------------------------
__global__ void k_gemm1(const float* __restrict__ h, const float* __restrict__ W1,
                        const float* __restrict__ b1, float* __restrict__ h1,
                        float* __restrict__ colsum, float* __restrict__ colsumsq) {
    int wave    = blockIdx.x * 8 + (threadIdx.x >> 5);
    int tileRow = wave >> 3;          // 3125 row tiles * 16 rows = 50000
    int tileCol = wave & 7;           // 8 col tiles * 16 = 128
    int lane    = threadIdx.x & 31;
    int half    = lane >> 4;          // 0: K pair {0,1}; 1: K pair {2,3}
    int l16     = lane & 15;

    int row = tileRow * 16 + l16;
    int n   = tileCol * 16 + l16;

    const float* arow = h  + (size_t)row * DIM + half * 2;
    const float* bcol = W1 + (size_t)(half * 2) * DIM + n;

    v8f c = {};
#pragma unroll 8
    for (int k = 0; k < DIM; k += 4) {
        v2f a = *(const v2f*)(arow + k);
        v2f b;
        b.x = bcol[(size_t)k * DIM];
        b.y = bcol[(size_t)k * DIM + DIM];
        c = __builtin_amdgcn_wmma_f32_16x16x4_f32(false, a, false, b,
                                                  (short)0, c, false, false);
    }

    float bias = b1[n];
    float s = 0.0f, sq = 0.0f;
    int rowBase = tileRow * 16 + half * 8;
#pragma unroll
    for (int r = 0; r < 8; ++r) {
        float v = c[r] + bias;
        h1[(size_t)(rowBase + r) * DIM + n] = v;
        s  += v;
        sq += v * v;
    }
    // lane l and l+16 hold the same column n -> combine, lanes 0-15 commit
    s  += __shfl_down(s, 16, 32);
    sq += __shfl_down(sq, 16, 32);
    if (lane < 16) {
        unsafeAtomicAdd(&colsum[n], s);
        unsafeAtomicAdd(&colsumsq[n], sq);
    }
}

// ---------------------------------------------------------------------------
// Kernel 5: fold BN stats into per-channel scale/shift (128 threads)
// ---------------------------------------------------------------------------
__global__ void k_bn_prep(const float* __restrict__ colsum,
                          const float* __restrict__ colsumsq,
                          const float* __restrict__ gamma,
                          const float* __restrict__ beta,
                          float* __restrict__ scale, float* __restrict__ shift) {
    int d = threadIdx.x;
    const float invN = 1.0f / (float)NNODES;
    float mean = colsum[d] * invN;
    float var  = colsumsq[d] * invN - mean * mean;
    float a    = gamma[d] * rsqrtf(var + BN_EPS);
    scale[d] = a;
    shift[d] = beta[d] - mean * a;
}

// ---------------------------------------------------------------------------
// Kernel 6: out = relu(h1*scale + shift) @ W2 + b2, WMMA f32, BN fused on A-load
// ---------------------------------------------------------------------------
__global__ void k_gemm2(const float* __restrict__ h1, const float* __restrict__ W2,
                        const float* __restrict__ b2,
                        const float* __restrict__ scale,
                        const float* __restrict__ shift,
                        float* __restrict__ out) {
    int wave    = blockIdx.x * 8 + (threadIdx.x >> 5);
    int tileRow = wave >> 3;
    int tileCol = wave & 7;
    int lane    = threadIdx.x & 31;
    int half    = lane >> 4;
    int l16     = lane & 15;

    int row = tileRow * 16 + l16;
    int n   = tileCol * 16 + l16;

    const float* arow = h1 + (size_t)row * DIM + half * 2;
    const float* bcol = W2 + (size_t)(half * 2) * DIM + n;

    v8f c = {};
#pragma unroll 8
    for (int k = 0; k < DIM; k += 4) {
        int kk = k + half * 2;
        v2f raw = *(const v2f*)(arow + k);
        v2f a;
        a.x = fmaxf(fmaf(raw.x, scale[kk],     shift[kk]),     0.0f);
        a.y = fmaxf(fmaf(raw.y, scale[kk + 1], shift[kk + 1]), 0.0f);
        v2f b;
        b.x = bcol[(size_t)k * DIM];
        b.y = bcol[(size_t)k * DIM + DIM];
        c = __builtin_amdgcn_wmma_f32_16x16x4_f32(false, a, false, b,
                                                  (short)0, c, false, false);
    }

    float bias = b2[n];
    int rowBase = tileRow * 16 + half * 8;
#pragma unroll
    for (int r = 0; r < 8; ++r) {
        out[(size_t)(rowBase + r) * DIM + n] = c[r] + bias;
    }
}

// ---------------------------------------------------------------------------
extern "C" void kernel_launch(void* const* d_in, const int* in_sizes, int n_in,
                              void* d_out, int out_size, void* d_ws, size_t ws_size,
                              hipStream_t stream) {
    const float* x         = (const float*)d_in[0];
    const int*   edge_attr = (const int*)d_in[1];
    const int*   src       = (const int*)d_in[2];
    const int*   dst       = (const int*)d_in[3];
    const float* bond_emb  = (const float*)d_in[4];
    const float* eps       = (const float*)d_in[5];
    const float* W1        = (const float*)d_in[6];
    const float* b1        = (const float*)d_in[7];
    const float* gamma     = (const float*)d_in[8];
    const float* beta      = (const float*)d_in[9];
    const float* W2        = (const float*)d_in[10];
    const float* b2        = (const float*)d_in[11];
    float* out = (float*)d_out;

    // workspace layout (floats)
    float* hbuf     = (float*)d_ws;                       // N*D
    float* h1       = hbuf + (size_t)NNODES * DIM;        // N*D
    float* stats    = h1   + (size_t)NNODES * DIM;        // 256 (sum|sumsq)
    float* colsum   = stats;
    float* colsumsq = stats + DIM;
    float* scale    = stats + 2 * DIM;                    // 128
    float* shift    = stats + 3 * DIM;                    // 128

    // 1) h = (1+eps) * x
    k_init_h<<<(NNODES * DIM / 4 + 255) / 256, 256, 0, stream>>>(x, eps, hbuf);

    // 2) edge messages + scatter-add (8 edges per 256-thread block)
    k_edge_scatter<<<(NEDGES + 7) / 8, 256, 0, stream>>>(x, edge_attr, src, dst,
                                                         bond_emb, hbuf);

    // 3) zero BN stats accumulators
    k_zero_stats<<<1, 256, 0, stream>>>(stats);

    // 4) GEMM1 + bias + column stats   (25000 tiles, 8 waves/block)
    k_gemm1<<<(NNODES / 16) * 8 / 8, 256, 0, stream>>>(hbuf, W1, b1, h1,
                                                       colsum, colsumsq);

    // 5) BN scale/shift
    k_bn_prep<<<1, DIM, 0, stream>>>(colsum, colsumsq, gamma, beta, scale, shift);

    // 6) GEMM2 with fused BN+ReLU on inputs
    k_gemm2<<<(NNODES / 16) * 8 / 8, 256, 0, stream>>>(h1, W2, b2, scale, shift, out);
}